// SecondOrderObjectiveConstraintComposition_67851893342498
// MI455X (gfx1250) — compile-verified
//
#include <hip/hip_runtime.h>
#include <cstdint>

#define N_DIM 128
#define M_DIM 32
#define LDJF  132   // padded row stride (floats) for JF in LDS: 132*4=528B, 16B aligned, bank-spread
#define LDV   33    // padded row stride for V
#define LDG   33    // padded row stride for 32x33 augmented system

typedef __attribute__((ext_vector_type(2))) float v2f;
typedef __attribute__((ext_vector_type(8))) float v8f;
typedef unsigned int v4u __attribute__((ext_vector_type(4)));
typedef int v4i __attribute__((ext_vector_type(4)));
typedef int v8i __attribute__((ext_vector_type(8)));

static __device__ __forceinline__ float wave_sum(float v) {
  v += __shfl_xor(v, 16, 32);
  v += __shfl_xor(v, 8, 32);
  v += __shfl_xor(v, 4, 32);
  v += __shfl_xor(v, 2, 32);
  v += __shfl_xor(v, 1, 32);
  return v;
}

// One-wave 32x32 = (32xK)(Kx32) GEMM via V_WMMA_F32_16X16X4_F32.
// A-operand: row-major [32][lda]. B-operand:
//   BTRANS=false: B[k][n] = Bm[k*ldb + n]   (Bm is K x 32)
//   BTRANS=true : B[k][n] = Bm[n*ldb + k]   (Bm is 32 x K, i.e. B = Bm^T)
template <bool BTRANS>
static __device__ __forceinline__ void wave_gemm32(const float* Am, int lda,
                                                   const float* Bm, int ldb,
                                                   float* G, int ldg, int K,
                                                   int lane) {
  const int l16 = lane & 15;
  const int h   = lane >> 4;   // 0: K={0,1}, 1: K={2,3} within each 4-wide step
#pragma unroll
  for (int m0 = 0; m0 < 32; m0 += 16) {
#pragma unroll
    for (int n0 = 0; n0 < 32; n0 += 16) {
      v8f acc = {0.f, 0.f, 0.f, 0.f, 0.f, 0.f, 0.f, 0.f};
      for (int k0 = 0; k0 < K; k0 += 4) {
        v2f a, b;
        a.x = Am[(m0 + l16) * lda + k0 + 2 * h + 0];
        a.y = Am[(m0 + l16) * lda + k0 + 2 * h + 1];
        if (BTRANS) {
          b.x = Bm[(n0 + l16) * ldb + k0 + 2 * h + 0];
          b.y = Bm[(n0 + l16) * ldb + k0 + 2 * h + 1];
        } else {
          b.x = Bm[(k0 + 2 * h + 0) * ldb + n0 + l16];
          b.y = Bm[(k0 + 2 * h + 1) * ldb + n0 + l16];
        }
        // (neg_a, A, neg_b, B, c_mod, C, reuse_a, reuse_b)
        acc = __builtin_amdgcn_wmma_f32_16x16x4_f32(false, a, false, b,
                                                    (short)0, acc, false, false);
      }
#pragma unroll
      for (int r = 0; r < 8; ++r)
        G[(m0 + r + 8 * h) * ldg + (n0 + l16)] = acc[r];
    }
  }
}

// One-wave in-place solve of the augmented system G[32][LDG] (col 32 = rhs).
// Gaussian elimination without pivoting (G is SPD here). Result in w[32].
static __device__ __forceinline__ void wave_solve32(float* G, float* w, int lane) {
  for (int k = 0; k < 31; ++k) {
    float piv = G[k * LDG + k];
    float f = G[lane * LDG + k] / piv;
    if (lane > k) {
      for (int c = k + 1; c <= 32; ++c)
        G[lane * LDG + c] -= f * G[k * LDG + c];
    }
    __syncthreads();  // single-wave WG: s_barrier is a NOP, acts as mem fence
  }
  for (int k = 31; k >= 0; --k) {
    float wk = G[k * LDG + 32] / G[k * LDG + k];
    if (lane < k) G[lane * LDG + 32] -= G[lane * LDG + k] * wk;
    if (lane == 0) w[k] = wk;
    __syncthreads();
  }
}

__global__ __launch_bounds__(32)
void SecondOrderObjectiveConstraintComposition_kernel(
    const float* __restrict__ X, const float* __restrict__ P,
    const float* __restrict__ A, float* __restrict__ OUT, int batch) {
  __shared__ float sJF[M_DIM][LDJF];   // JF, padded rows
  __shared__ float sV[N_DIM][LDV];     // V = Md^{-1} JF^T
  __shared__ float sG[M_DIM * LDG];    // augmented 32x33 system
  __shared__ float sW[M_DIM];          // solve result (w, then lambda)
  __shared__ float sQ[N_DIM];          // q = D^{-1} c
  __shared__ float sT[M_DIM];          // t_i = q . JF_row_i
  __shared__ float sU[N_DIM];          // u = Md^{-1} rhs

  const int item = blockIdx.x;
  if (item >= batch) return;
  const int lane = threadIdx.x;
  const float* x = X + (size_t)item * N_DIM;
  const float* p = P + (size_t)item * N_DIM;

  // ---- Stage A (32x128 f32) into LDS with 132-float padded rows ----
#if __has_builtin(__builtin_amdgcn_tensor_load_to_lds)
  {
    uint64_t ga = (uint64_t)(uintptr_t)A;
    uint32_t lds_addr = (uint32_t)(uintptr_t)(&sJF[0][0]);
    v4u g0;
    g0[0] = 1u;                                  // count=1, user descriptor
    g0[1] = lds_addr;                            // lds_addr
    g0[2] = (uint32_t)ga;                        // global_addr[31:0]
    g0[3] = (uint32_t)((ga >> 32) & 0x01FFFFFFu) // global_addr[56:32]
            | 0x80000000u;                       // type=2 ("image") in bits 127:126
    v8i g1;
    // data_size=2 (4B), pad_enable=1, pad_interval=6 (128 DWORDs), pad_amount=3 (4 DWORDs)
    g1[0] = (int)((2u << 16) | (1u << 20) | (6u << 22) | (3u << 25));
    g1[1] = (int)(128u << 16);   // tensor_dim0 = 128
    g1[2] = (int)(32u << 16);    // tensor_dim1 = 32
    g1[3] = (int)(128u << 16);   // tile_dim0 = 128
    g1[4] = 32;                  // tile_dim1 = 32 (tile_dim2 = 0)
    g1[5] = 128;                 // tensor_dim0_stride = 128
    g1[6] = 0;
    g1[7] = 0;
    v4i g2 = {0, 0, 0, 0}, g3 = {0, 0, 0, 0};
#if __clang_major__ >= 23
    v8i g4 = {0, 0, 0, 0, 0, 0, 0, 0};
    __builtin_amdgcn_tensor_load_to_lds(g0, g1, g2, g3, g4, 0);
#else
    __builtin_amdgcn_tensor_load_to_lds(g0, g1, g2, g3, 0);
#endif
#if __has_builtin(__builtin_amdgcn_s_wait_tensorcnt)
    __builtin_amdgcn_s_wait_tensorcnt(0);
#endif
  }
#else
  for (int r = 0; r < M_DIM; ++r) {
    float4 v = *(const float4*)(A + r * N_DIM + lane * 4);
    *(float4*)(&sJF[r][lane * 4]) = v;
  }
#endif

  // ---- Per-element quantities; lane owns j = lane + 32k, k=0..3 ----
  float xv[4], pv[4], sn[4], cs[4], di[4], dg[4], fg[4], qv[4];
  float s_loc = 0.f;
#pragma unroll
  for (int k = 0; k < 4; ++k) {
    int j = lane + 32 * k;
    xv[k] = x[j];
    pv[k] = p[j];
    sn[k] = sinf(xv[k]);
    cs[k] = cosf(xv[k]);
    s_loc += sn[k];
  }
  const float s = wave_sum(s_loc);  // s = sum(sin x)
  float bq = 0.f;
#pragma unroll
  for (int k = 0; k < 4; ++k) {
    int j = lane + 32 * k;
    float p2 = pv[k] * pv[k];
    dg[k] = 1.f + p2 - 0.02f * s * sn[k];        // D (gamma/2 = 1)
    di[k] = 1.f / dg[k];                         // D^{-1}
    fg[k] = xv[k] * p2 + 0.02f * s * cs[k];      // grad f
    qv[k] = di[k] * cs[k];                       // q = D^{-1} c
    sQ[j] = qv[k];
    bq += cs[k] * qv[k];                         // c^T D^{-1} c partial
  }
  const float beta  = wave_sum(bq);
  const float alpha = 0.02f / (1.f + 0.02f * beta);  // Sherman-Morrison factor

  // b_i = p_i + 0.1*(x_i*sech^2 - tanh)  (the A*x terms cancel in -F + JF*x)
  const float th = tanhf(xv[0]);
  const float sech2 = 1.f - th * th;
  const float ei = 0.1f * sech2;
  const float b_i = pv[0] + 0.1f * (xv[0] * sech2 - th);

  __syncthreads();
  sJF[lane][lane] += ei;   // JF = A + 0.1*diag(sech^2) on first 32 cols
  __syncthreads();

  // ---- G1 = JF JF^T (WMMA), solve G1 w = b ----
  wave_gemm32<true>(&sJF[0][0], LDJF, &sJF[0][0], LDJF, sG, LDG, N_DIM, lane);
  __syncthreads();
  sG[lane * LDG + 32] = b_i;
  __syncthreads();
  wave_solve32(sG, sW, lane);
  __syncthreads();

  // ---- zeta = JF^T w ; rhs = Md(x - zeta) - fg ; u = Md^{-1} rhs ----
  float ze[4], xmz[4];
#pragma unroll
  for (int k = 0; k < 4; ++k) {
    int j = lane + 32 * k;
    float acc = 0.f;
    for (int i = 0; i < M_DIM; ++i) acc += sJF[i][j] * sW[i];
    ze[k]  = acc;
    xmz[k] = xv[k] - acc;
  }
  const float dot1 = wave_sum(cs[0]*xmz[0] + cs[1]*xmz[1] + cs[2]*xmz[2] + cs[3]*xmz[3]);
  float rhs[4], d2 = 0.f;
#pragma unroll
  for (int k = 0; k < 4; ++k) {
    rhs[k] = dg[k] * xmz[k] + 0.02f * cs[k] * dot1 - fg[k];  // Md(x-z) - fg
    d2 += qv[k] * rhs[k];
  }
  const float dot2 = wave_sum(d2);
  float uu[4];
#pragma unroll
  for (int k = 0; k < 4; ++k) {
    int j = lane + 32 * k;
    uu[k] = di[k] * rhs[k] - alpha * qv[k] * dot2;  // u = Md^{-1} rhs
    sU[j] = uu[k];
  }
  __syncthreads();

  // ---- t_i = q . JF_i ; r2_i = JF_i . u ----
  float t = 0.f, r2 = 0.f;
  for (int j = 0; j < N_DIM; ++j) {
    float jf = sJF[lane][j];
    t  += sQ[j] * jf;
    r2 += sU[j] * jf;
  }
  sT[lane] = t;
  __syncthreads();

  // ---- V = Md^{-1} JF^T : V[j][i] = di_j*JF[i][j] - alpha*q_j*t_i ----
#pragma unroll
  for (int k = 0; k < 4; ++k) {
    int j = lane + 32 * k;
    float dij = di[k], qj = qv[k];
    for (int i = 0; i < M_DIM; ++i)
      sV[j][i] = dij * sJF[i][j] - alpha * qj * sT[i];
  }
  __syncthreads();

  // ---- G2 = JF V (WMMA), solve G2 lambda = JF u ----
  wave_gemm32<false>(&sJF[0][0], LDJF, &sV[0][0], LDV, sG, LDG, N_DIM, lane);
  __syncthreads();
  sG[lane * LDG + 32] = r2;
  __syncthreads();
  wave_solve32(sG, sW, lane);   // sW = lambda
  __syncthreads();

  // ---- x_new = zeta + u - V lambda ----
#pragma unroll
  for (int k = 0; k < 4; ++k) {
    int j = lane + 32 * k;
    float acc = 0.f;
    for (int i = 0; i < M_DIM; ++i) acc += sV[j][i] * sW[i];
    OUT[(size_t)item * N_DIM + j] = ze[k] + uu[k] - acc;
  }
}

extern "C" void kernel_launch(void* const* d_in, const int* in_sizes, int n_in,
                              void* d_out, int out_size, void* d_ws, size_t ws_size,
                              hipStream_t stream) {
  const float* x = (const float*)d_in[0];
  const float* p = (const float*)d_in[1];
  const float* A = (const float*)d_in[2];
  float* out = (float*)d_out;
  const int batch = in_sizes[0] / N_DIM;
  SecondOrderObjectiveConstraintComposition_kernel
      <<<dim3(batch), dim3(32), 0, stream>>>(x, p, A, out, batch);
}